// PhysicalTokenizer_13907104104849
// MI455X (gfx1250) — compile-verified
//
#include <hip/hip_runtime.h>
#include <cstdint>

typedef unsigned int v4u __attribute__((ext_vector_type(4)));
typedef int          v8i __attribute__((ext_vector_type(8)));
typedef int          v4i __attribute__((ext_vector_type(4)));
typedef float        v4f __attribute__((ext_vector_type(4)));

#define PI_F 3.14159265358979323846f

constexpr int S_LEN   = 8192;
constexpr int EMBED   = 128;
constexpr int NCHARS  = 95;
constexpr int PPARAM  = 8;
constexpr int W_ELEMS = NCHARS * PPARAM;      // 760 floats = 3040 bytes
constexpr int TOK_PER_BLOCK = 2;
constexpr int BLOCK   = TOK_PER_BLOCK * EMBED; // 256 threads = 8 wave32

__global__ __launch_bounds__(BLOCK)
void phys_tok_kernel(const int* __restrict__ char_idx,
                     const int* __restrict__ positions,
                     const float* __restrict__ W,
                     float* __restrict__ out)
{
    __shared__ float sW[W_ELEMS];
    __shared__ float sWave[TOK_PER_BLOCK * EMBED];

    // ---- Stage W (95x8 f32, 3040 B) into LDS via the Tensor Data Mover ----
    {
        const uint64_t gaddr   = (uint64_t)(uintptr_t)W;
        const uint32_t lds_off = (uint32_t)(uintptr_t)&sW[0]; // low 32 bits of LDS aperture = LDS byte addr

        v4u g0;
        g0[0] = 1u;                                   // count = 1 valid descriptor
        g0[1] = lds_off;                              // lds_addr (bytes)
        g0[2] = (uint32_t)(gaddr & 0xFFFFFFFFu);      // global_addr[31:0]
        g0[3] = ((uint32_t)(gaddr >> 32) & 0x01FFFFFFu) | (2u << 30); // addr[56:32] | type=2

        v8i g1;
        g1[0] = (int)(2u << 16);                      // data_size = 2 (4 bytes/elem)
        g1[1] = (int)((uint32_t)W_ELEMS << 16);       // tensor_dim0[15:0] in bits[63:48]
        g1[2] = (int)(1u << 16);                      // tensor_dim0 hi = 0; tensor_dim1 = 1
        g1[3] = (int)((uint32_t)W_ELEMS << 16);       // tile_dim0 = 760 in bits[127:112]
        g1[4] = 0;                                    // tile_dim1 = tile_dim2 = 0 (unused)
        g1[5] = W_ELEMS;                              // tensor_dim0_stride lo
        g1[6] = 0;
        g1[7] = 0;

        v4i z4 = {0, 0, 0, 0};                        // <=2-D tensor: groups 2/3 unused
        v8i z8 = {0, 0, 0, 0, 0, 0, 0, 0};            // trailing group (6-arg builtin form)

        if (threadIdx.x < 32u) {                      // one DMA per block (wave 0)
            __builtin_amdgcn_tensor_load_to_lds(g0, g1, z4, z4, z8, 0);
            __builtin_amdgcn_s_wait_tensorcnt(0);     // TENSORcnt == 0 before sharing
        }
    }
    __syncthreads();

    // ---- Per-token / per-j wave synthesis ----
    const uint32_t tid = threadIdx.x;
    const int tib   = (int)(tid >> 7);                // token within block
    const int j     = (int)(tid & (EMBED - 1));       // embed index
    const int token = (int)blockIdx.x * TOK_PER_BLOCK + tib;
    const int s     = token & (S_LEN - 1);

    const int    c = char_idx[token];
    const float* p = &sW[c * PPARAM];

    const float omega = p[0] * 2.0f;
    const float A1 = p[1], A2 = p[2], A3 = p[3];
    const float beta  = p[4];
    const float gamma = __builtin_amdgcn_rcpf(1.0f + __expf(-p[5])); // sigmoid
    const float phi   = p[6] * PI_F;

    const float jf  = (float)j;
    const float arg = omega * jf + phi;               // k*omega*j + k*phi == k*arg
    const float env = __expf(-gamma * jf);

    float wave = (A1 * __sinf(arg)
                + A2 * __sinf(2.0f * arg)
                + A3 * __sinf(3.0f * arg)) * env;

    const float posf = (float)positions[s];
    wave += beta * jf * __sinf(posf * (0.1f * PI_F));

    // roll(wave, 1) along j via LDS
    sWave[tid] = wave;
    __syncthreads();
    const float psi1 = sWave[(tib << 7) | ((j + EMBED - 1) & (EMBED - 1))];

    v4f o;
    o[0] = wave;
    o[1] = psi1;
    o[2] = __sinf(wave);
    o[3] = __cosf(wave);

    // [B,S,128,4]: one float4 per (token, j) -> coalesced b128 NT store stream
    __builtin_nontemporal_store(o, (v4f*)out + (size_t)token * EMBED + j);
}

extern "C" void kernel_launch(void* const* d_in, const int* in_sizes, int n_in,
                              void* d_out, int out_size, void* d_ws, size_t ws_size,
                              hipStream_t stream) {
    const int*   char_idx  = (const int*)d_in[0];   // [B*S] int32
    const int*   positions = (const int*)d_in[1];   // [S]   int32
    const float* W         = (const float*)d_in[2]; // [95*8] f32
    float*       out       = (float*)d_out;         // [B*S*128*4] f32

    const int n_tokens = in_sizes[0];               // 65536
    const int grid     = (n_tokens + TOK_PER_BLOCK - 1) / TOK_PER_BLOCK;

    phys_tok_kernel<<<dim3(grid), dim3(BLOCK), 0, stream>>>(char_idx, positions, W, out);
}